// GCNModule_7121055776908
// MI455X (gfx1250) — compile-verified
//
#include <hip/hip_runtime.h>
#include <hip/hip_bf16.h>

typedef __attribute__((ext_vector_type(2))) float v2f;
typedef __attribute__((ext_vector_type(8))) float v8f;

#define D_FEAT 64
#define GEMM_WAVES 8   // waves (M-tiles) per block

// ---------------------------------------------------------------------------
// degree / normalization
// ---------------------------------------------------------------------------
__global__ void gcn_zero_f32(float* __restrict__ p, int n) {
    int i = blockIdx.x * blockDim.x + threadIdx.x;
    if (i < n) p[i] = 0.0f;
}

__global__ void gcn_degree(const int* __restrict__ dst, float* __restrict__ deg, int E) {
    int i = blockIdx.x * blockDim.x + threadIdx.x;
    if (i < E) atomicAdd(&deg[dst[i]], 1.0f);
}

__global__ void gcn_dinv(const float* __restrict__ deg, float* __restrict__ dinv, int N) {
    int i = blockIdx.x * blockDim.x + threadIdx.x;
    if (i < N) dinv[i] = __frsqrt_rn(deg[i] + 1.0f);   // +1 = self loop; always > 0
}

// ---------------------------------------------------------------------------
// H = X @ W  via V_WMMA_F32_16X16X4_F32.
// 8 waves per block; each wave computes one 16x64 strip (4 f32 accumulators).
// W is staged in LDS PAIR-MAJOR: Wl[kp][n] holds (W[2kp][n], W[2kp+1][n]) as
// an adjacent float pair, so each B fragment is one aligned ds_load_b64
// directly into an even VGPR pair (no repack movs).
// A 16x4 layout: lane<16 -> M=lane, K={k0,k0+1}; lane>=16 -> K={k0+2,k0+3}
// C/D: VGPR v, lane -> M = v + (lane>>4)*8, N = lane&15
// ---------------------------------------------------------------------------
__global__ __launch_bounds__(32 * GEMM_WAVES) void gcn_gemm_wmma(
        const float* __restrict__ X, const float* __restrict__ W,
        float* __restrict__ H, int M) {
    __shared__ float Wl[2 * 32 * D_FEAT];          // [kp=32][n=64][pair=2] = 16KB
    const int tid  = threadIdx.x;
    const int lane = tid & 31;
    const int wave = tid >> 5;

    // stage W pair-major
    for (int idx = tid; idx < 32 * D_FEAT; idx += 32 * GEMM_WAVES) {
        int kp = idx >> 6;
        int n  = idx & 63;
        v2f w;
        w.x = W[(2 * kp)     * D_FEAT + n];
        w.y = W[(2 * kp + 1) * D_FEAT + n];
        *(v2f*)&Wl[idx * 2] = w;                   // 8B-aligned ds_store_b64
    }
    __syncthreads();

    const int mt = blockIdx.x * GEMM_WAVES + wave;
    const int m0 = mt * 16;
    if (m0 >= M) return;                           // after the only barrier: safe

    const int row  = lane & 15;                    // M within tile (A) / N within tile (B)
    const int kodd = lane >> 4;                    // selects K pair {0,1} vs {2,3}

    v8f c0 = {}, c1 = {}, c2 = {}, c3 = {};
    const float* xrow = X + (size_t)(m0 + row) * D_FEAT;

    #pragma unroll
    for (int k0 = 0; k0 < D_FEAT; k0 += 4) {
        v2f a = *(const v2f*)(xrow + k0 + 2 * kodd);        // global_load_b64

        const int    kp = (k0 >> 1) + kodd;                 // K-pair index
        const float* wb = Wl + (size_t)kp * (2 * D_FEAT);
        v2f b0 = *(const v2f*)(wb + (row)      * 2);        // ds_load_b64 each
        v2f b1 = *(const v2f*)(wb + (16 + row) * 2);
        v2f b2 = *(const v2f*)(wb + (32 + row) * 2);
        v2f b3 = *(const v2f*)(wb + (48 + row) * 2);

        c0 = __builtin_amdgcn_wmma_f32_16x16x4_f32(false, a, false, b0, (short)0, c0, false, false);
        c1 = __builtin_amdgcn_wmma_f32_16x16x4_f32(false, a, false, b1, (short)0, c1, false, false);
        c2 = __builtin_amdgcn_wmma_f32_16x16x4_f32(false, a, false, b2, (short)0, c2, false, false);
        c3 = __builtin_amdgcn_wmma_f32_16x16x4_f32(false, a, false, b3, (short)0, c3, false, false);
    }

    const int srow = (lane >> 4) * 8;
    const int scol = lane & 15;
    #pragma unroll
    for (int v = 0; v < 8; ++v) {
        int m = m0 + srow + v;
        if (m < M) {
            float* hr = H + (size_t)m * D_FEAT + scol;
            hr[ 0] = c0[v];
            hr[16] = c1[v];
            hr[32] = c2[v];
            hr[48] = c3[v];
        }
    }
}

// ---------------------------------------------------------------------------
// acc[i] = H[i] * dinv[i]^2   (self-loop contribution, also zero-initializes acc)
// ---------------------------------------------------------------------------
__global__ void gcn_selfloop(const float* __restrict__ H, const float* __restrict__ dinv,
                             float* __restrict__ acc, int total) {
    int t = blockIdx.x * blockDim.x + threadIdx.x;
    if (t >= total) return;
    int node = t >> 6;
    float di = dinv[node];
    acc[t] = H[t] * di * di;
}

// ---------------------------------------------------------------------------
// edge scatter: acc[dst] += H[src] * dinv[src]*dinv[dst]
// 16 threads per edge, float4 per thread; atomics resolve in the 192MB L2.
// ---------------------------------------------------------------------------
__global__ void gcn_scatter(const int* __restrict__ src, const int* __restrict__ dst,
                            const float* __restrict__ dinv, const float* __restrict__ H,
                            float* __restrict__ acc, int E) {
    int t = blockIdx.x * blockDim.x + threadIdx.x;
    int e = t >> 4;
    if (e >= E) return;
    int part = (t & 15) * 4;
    int s = src[e];
    int d = dst[e];
    // hint the next edge's gather row toward the caches (global_prefetch_b8)
    if (e + 1 < E) __builtin_prefetch(H + (size_t)src[e + 1] * D_FEAT + part, 0, 3);
    float norm = dinv[s] * dinv[d];
    float4 hv = *(const float4*)(H + (size_t)s * D_FEAT + part);
    float* out = acc + (size_t)d * D_FEAT + part;
    atomicAdd(out + 0, hv.x * norm);
    atomicAdd(out + 1, hv.y * norm);
    atomicAdd(out + 2, hv.z * norm);
    atomicAdd(out + 3, hv.w * norm);
}

// ---------------------------------------------------------------------------
// x_next = sigmoid(acc + b)
// ---------------------------------------------------------------------------
__global__ void gcn_finalize(const float* __restrict__ acc, const float* __restrict__ b,
                             float* __restrict__ xnext, int total) {
    int t = blockIdx.x * blockDim.x + threadIdx.x;
    if (t >= total) return;
    float v = acc[t] + b[t & 63];
    xnext[t] = 1.0f / (1.0f + __expf(-v));
}

// ---------------------------------------------------------------------------
// out[a, layer*64 + d] = x[agent_idx[a], d]
// ---------------------------------------------------------------------------
__global__ void gcn_gather_agents(const float* __restrict__ x, const int* __restrict__ agent_idx,
                                  float* __restrict__ out, int nA, int layer) {
    int t = blockIdx.x * blockDim.x + threadIdx.x;
    if (t >= nA * D_FEAT) return;
    int a = t >> 6;
    int dcol = t & 63;
    int node = agent_idx[a];
    out[(size_t)a * (3 * D_FEAT) + layer * D_FEAT + dcol] = x[(size_t)node * D_FEAT + dcol];
}

// ---------------------------------------------------------------------------
extern "C" void kernel_launch(void* const* d_in, const int* in_sizes, int n_in,
                              void* d_out, int out_size, void* d_ws, size_t ws_size,
                              hipStream_t stream) {
    const float* x0    = (const float*)d_in[0];
    const int*   edge  = (const int*)d_in[1];       // [2, E] flat: src then dst
    const int*   agent = (const int*)d_in[3];
    const float* Wm[3] = {(const float*)d_in[4], (const float*)d_in[6], (const float*)d_in[8]};
    const float* bm[3] = {(const float*)d_in[5], (const float*)d_in[7], (const float*)d_in[9]};

    const int N  = in_sizes[0] / D_FEAT;            // 50000
    const int E  = in_sizes[1] / 2;                 // 800000
    const int nA = in_sizes[3];                     // 12500
    const int ND = N * D_FEAT;

    const int* src = edge;
    const int* dst = edge + E;

    // workspace carve-up (all f32): deg[N] | dinv[N] | xbuf[N*64] | h[N*64] | acc[N*64]
    float* ws   = (float*)d_ws;
    float* deg  = ws;
    float* dinv = deg + N;
    float* xbuf = dinv + N;
    float* h    = xbuf + ND;
    float* acc  = h + ND;

    const int B = 256;
    const int Mtiles = (N + 15) / 16;

    // normalization (recomputed deterministically every call; ws is poisoned)
    gcn_zero_f32<<<(N + B - 1) / B, B, 0, stream>>>(deg, N);
    gcn_degree <<<(E + B - 1) / B, B, 0, stream>>>(dst, deg, E);
    gcn_dinv   <<<(N + B - 1) / B, B, 0, stream>>>(deg, dinv, N);

    const float* xin = x0;
    for (int l = 0; l < 3; ++l) {
        gcn_gemm_wmma<<<(Mtiles + GEMM_WAVES - 1) / GEMM_WAVES, 32 * GEMM_WAVES, 0, stream>>>(
            xin, Wm[l], h, N);
        gcn_selfloop <<<(ND + B - 1) / B, B, 0, stream>>>(h, dinv, acc, ND);
        gcn_scatter  <<<((size_t)E * 16 + B - 1) / B, B, 0, stream>>>(src, dst, dinv, h, acc, E);
        gcn_finalize <<<(ND + B - 1) / B, B, 0, stream>>>(acc, bm[l], xbuf, ND);
        gcn_gather_agents<<<(nA * D_FEAT + B - 1) / B, B, 0, stream>>>(xbuf, agent, (float*)d_out, nA, l);
        xin = xbuf;
    }
}